// FreqPSR_16810501997327
// MI455X (gfx1250) — compile-verified
//
#include <hip/hip_runtime.h>
#include <math.h>

typedef __attribute__((ext_vector_type(2))) float v2f;
typedef __attribute__((ext_vector_type(8))) float v8f;
typedef __attribute__((__vector_size__(4 * sizeof(int)))) int vi4;

#define NP 64
#define NCH 3
#define NE 16
#define NELEM (NCH * NP * NE)  // 3072

// D = A(16x4) * B(4x16) + C  on the f32 WMMA pipe; chained over 4 K-chunks
// to realize the 16x16 complex 2D-IDFT matrix.
__global__ __launch_bounds__(256) void freq_psr_kernel(
    const float* __restrict__ adv, const float* __restrict__ cle,
    float* __restrict__ out, int Bn) {
  __shared__ float sCpha[NELEM];   // clean phase (equality compare, phase attack)
  __shared__ float sCamp[NELEM];   // clean amplitude (equality compare, amp attack)
  __shared__ float sFr[NELEM];     // f_cle real   (un-swapped reconstruction value)
  __shared__ float sFi[NELEM];     // f_cle imag
  __shared__ float sPr[NELEM];     // C_amp * u_adv real (phase-swap value)
  __shared__ float sPi[NELEM];
  __shared__ float sQr[NELEM];     // A_amp * u_cle real (amp-swap value)
  __shared__ float sQi[NELEM];
  __shared__ float sAamp[NELEM];   // temp: adversarial amplitude
  __shared__ float sStage[8][1024];  // per-wave 32x32 output staging
  // The first 6144 floats of sStage double as the raw-image staging buffer
  // during passes 0/1 (sStage proper is only touched after the syncthreads).
  float* imgbuf = &sStage[0][0];  // [im*3072 + ch*1024 + row*32 + col]

  const int b = blockIdx.x;
  const int tid = threadIdx.x;
  const int lane = tid & 31;
  const int w = tid >> 5;

  // ---------- pass 0: stage both images into LDS (async DMA if available) ----
  {
    const float* gA = adv + (size_t)b * (NCH * 32 * 32);
    const float* gC = cle + (size_t)b * (NCH * 32 * 32);
#if __has_builtin(__builtin_amdgcn_global_load_async_to_lds_b128) && \
    __has_builtin(__builtin_amdgcn_s_wait_asynccnt)
    for (int c = tid; c < 384; c += 256) {  // 1536 floats = 384 b128 chunks
      const float* g = (c < 192) ? (gA + c * 4) : (gC + (c - 192) * 4);
      __builtin_amdgcn_global_load_async_to_lds_b128(
          (__attribute__((address_space(1))) vi4*)g,
          (__attribute__((address_space(3))) vi4*)&imgbuf[c * 4],
          /*offset=*/0, /*cpol=*/0);
    }
    __builtin_amdgcn_s_wait_asynccnt(0);
#else
    for (int c = tid; c < 384; c += 256) {
      const float4 v = (c < 192) ? *(const float4*)(gA + c * 4)
                                 : *(const float4*)(gC + (c - 192) * 4);
      *(float4*)&imgbuf[c * 4] = v;
    }
#endif
  }
  __syncthreads();

  // ---------- pass 1: forward 4x4 FFT of every patch of both images ----------
  for (int task = tid; task < 2 * NCH * NP; task += 256) {
    const int im = task / (NCH * NP);          // 0 = adversarial, 1 = clean
    const int rem = task - im * (NCH * NP);
    const int ch = rem >> 6;
    const int p = rem & 63;
    const int py = p >> 3, px = p & 7;
    const float* src = imgbuf + (im * NCH + ch) * 1024 + (py * 4) * 32 + px * 4;
    float x[4][4];
    for (int r = 0; r < 4; ++r) {
      const float4 v = *(const float4*)(src + r * 32);
      x[r][0] = v.x; x[r][1] = v.y; x[r][2] = v.z; x[r][3] = v.w;
    }
    // row DFT4 of real input: X[k] = sum x[n] * (-i)^(kn)
    float Rr[4][4], Ri[4][4];
    for (int r = 0; r < 4; ++r) {
      const float a0 = x[r][0], a1 = x[r][1], a2 = x[r][2], a3 = x[r][3];
      Rr[r][0] = a0 + a1 + a2 + a3; Ri[r][0] = 0.f;
      Rr[r][1] = a0 - a2;           Ri[r][1] = a3 - a1;
      Rr[r][2] = a0 - a1 + a2 - a3; Ri[r][2] = 0.f;
      Rr[r][3] = a0 - a2;           Ri[r][3] = a1 - a3;
    }
    // column DFT4 (complex)
    float Gr[4][4], Gi[4][4];
    for (int c = 0; c < 4; ++c) {
      const float r0 = Rr[0][c], i0 = Ri[0][c], r1 = Rr[1][c], i1 = Ri[1][c];
      const float r2 = Rr[2][c], i2 = Ri[2][c], r3 = Rr[3][c], i3 = Ri[3][c];
      const float t0r = r0 + r2, t0i = i0 + i2, t1r = r0 - r2, t1i = i0 - i2;
      const float t2r = r1 + r3, t2i = i1 + i3, t3r = r1 - r3, t3i = i1 - i3;
      Gr[0][c] = t0r + t2r; Gi[0][c] = t0i + t2i;
      Gr[2][c] = t0r - t2r; Gi[2][c] = t0i - t2i;
      Gr[1][c] = t1r + t3i; Gi[1][c] = t1i - t3r;  // t1 + (-i)*t3
      Gr[3][c] = t1r - t3i; Gi[3][c] = t1i + t3r;  // t1 + ( i)*t3
    }
    const int base = (ch * NP + p) * NE;
    for (int u = 0; u < 4; ++u)
      for (int v = 0; v < 4; ++v) {
        const float re = Gr[u][v], ii = Gi[u][v];
        const float amp = __builtin_amdgcn_sqrtf(re * re + ii * ii);
        float ur, ui;
        if (amp > 0.f) {
          const float inv = __builtin_amdgcn_rcpf(amp);
          ur = re * inv; ui = ii * inv;
        } else {
          ur = 1.f; ui = 0.f;  // exp(i*atan2(0,0)) = 1
        }
        const int o = base + u * 4 + v;
        if (im == 1) {
          sCpha[o] = atan2f(ii, re);
          sCamp[o] = amp;
          sFr[o] = re; sFi[o] = ii;
          sQr[o] = ur; sQi[o] = ui;  // u_cle, scaled by A_amp below
        } else {
          sAamp[o] = amp;
          sPr[o] = ur; sPi[o] = ui;  // u_adv, scaled by C_amp below
        }
      }
  }
  __syncthreads();
  for (int o = tid; o < NELEM; o += 256) {
    const float ca = sCamp[o], aa = sAamp[o];
    sPr[o] *= ca; sPi[o] *= ca;   // P = C_amp * u_adv
    sQr[o] *= aa; sQi[o] *= aa;   // Q = A_amp * u_cle
  }
  __syncthreads();

  // ---------- per-lane constant B operand: transposed 16x16 2D-IDFT ----------
  // B layout (4x16 f32): VGPR0 -> K = kc*4 + (lane<16 ? 0 : 2), VGPR1 -> K+1,
  // N = lane & 15.  Mt[k][j] = (1/16) * i^((jm*ku + jn*kv) & 3)
  const int n = lane & 15;
  const int jm = n >> 2, jn = n & 3;
  v2f btr[4], bti[4], btin[4];
  for (int kc = 0; kc < 4; ++kc) {
    const int k0 = kc * 4 + ((lane >> 4) << 1);
    float re[2], iv[2];
    for (int t = 0; t < 2; ++t) {
      const int k = k0 + t;
      const int e = (jm * (k >> 2) + jn * (k & 3)) & 3;
      re[t] = (e == 0) ? 0.0625f : ((e == 2) ? -0.0625f : 0.0f);
      iv[t] = (e == 1) ? 0.0625f : ((e == 3) ? -0.0625f : 0.0f);
    }
    btr[kc] = (v2f){re[0], re[1]};
    bti[kc] = (v2f){iv[0], iv[1]};
    btin[kc] = (v2f){-iv[0], -iv[1]};
  }

  float* stage = sStage[w];
  const size_t typeStride = (size_t)Bn * 64 * NCH * 1024;

  // ---------- pass 2: one (type, attack idx, channel) task per wave ----------
  for (int task = w; task < 2 * 64 * NCH; task += 8) {
    const int type = task / (64 * NCH);           // 0 = phase attack, 1 = amp
    const int rem = task - type * (64 * NCH);
    const int idx = rem / NCH;
    const int ch = rem - idx * NCH;

    for (int g = 0; g < 4; ++g) {                 // 16 patches per A-tile
      v8f Yr = {}; v8f Yi = {};
      const int p = g * 16 + (lane & 15);         // A: M = lane&15 (both halves)
      const int rowbase = (ch * NP + p) * NE;
      const int idxbase = (ch * NP + idx) * NE;
      const int koff = (lane >> 4) << 1;          // A: upper half holds K=2,3
      for (int kc = 0; kc < 4; ++kc) {
        const int kb = kc * 4 + koff;
        const int o = rowbase + kb, oi = idxbase + kb;
        v2f ar, ai;
        const v2f fr = *(const v2f*)&sFr[o], fi = *(const v2f*)&sFi[o];
        if (type == 0) {
          const v2f cp = *(const v2f*)&sCpha[o];
          const v2f cq = *(const v2f*)&sCpha[oi];
          const v2f pr = *(const v2f*)&sPr[o], pi = *(const v2f*)&sPi[o];
          ar.x = (cp.x == cq.x) ? pr.x : fr.x;
          ar.y = (cp.y == cq.y) ? pr.y : fr.y;
          ai.x = (cp.x == cq.x) ? pi.x : fi.x;
          ai.y = (cp.y == cq.y) ? pi.y : fi.y;
        } else {
          const v2f cp = *(const v2f*)&sCamp[o];
          const v2f cq = *(const v2f*)&sCamp[oi];
          const v2f qr = *(const v2f*)&sQr[o], qi = *(const v2f*)&sQi[o];
          ar.x = (cp.x == cq.x) ? qr.x : fr.x;
          ar.y = (cp.y == cq.y) ? qr.y : fr.y;
          ai.x = (cp.x == cq.x) ? qi.x : fi.x;
          ai.y = (cp.y == cq.y) ? qi.y : fi.y;
        }
        // Yr += Xr*Mtr - Xi*Mti ; Yi += Xr*Mti + Xi*Mtr
        Yr = __builtin_amdgcn_wmma_f32_16x16x4_f32(false, ar, false, btr[kc],
                                                   (short)0, Yr, false, false);
        Yr = __builtin_amdgcn_wmma_f32_16x16x4_f32(false, ai, false, btin[kc],
                                                   (short)0, Yr, false, false);
        Yi = __builtin_amdgcn_wmma_f32_16x16x4_f32(false, ar, false, bti[kc],
                                                   (short)0, Yi, false, false);
        Yi = __builtin_amdgcn_wmma_f32_16x16x4_f32(false, ai, false, btr[kc],
                                                   (short)0, Yi, false, false);
      }
      // |IFFT| per element; D layout: VGPR r -> row M = r (+8 for lanes>=16)
      const int m0 = (lane >> 4) << 3;
      for (int r = 0; r < 8; ++r) {
        const float yr = Yr[r], yi = Yi[r];
        const float mag = __builtin_amdgcn_sqrtf(yr * yr + yi * yi);
        const int pp = g * 16 + m0 + r;
        const int py = pp >> 3, px = pp & 7;
        stage[(py * 4 + jm) * 32 + (px * 4 + jn)] = mag;
      }
    }
    // coalesced b128 stream of the finished 32x32 image
    float* dst = out + type * typeStride + (((size_t)b * 64 + idx) * NCH + ch) * 1024;
    for (int c4 = 0; c4 < 8; ++c4) {
      const int fo = c4 * 128 + lane * 4;
      *(float4*)(dst + fo) = *(const float4*)&stage[fo];
    }
  }
}

extern "C" void kernel_launch(void* const* d_in, const int* in_sizes, int n_in,
                              void* d_out, int out_size, void* d_ws, size_t ws_size,
                              hipStream_t stream) {
  (void)n_in; (void)out_size; (void)d_ws; (void)ws_size;
  const float* adv = (const float*)d_in[0];
  const float* cle = (const float*)d_in[1];
  float* out = (float*)d_out;
  const int B = in_sizes[0] / (3 * 32 * 32);  // 128
  freq_psr_kernel<<<dim3(B), dim3(256), 0, stream>>>(adv, cle, out, B);
}